// Cell_5377299054722
// MI455X (gfx1250) — compile-verified
//
#include <hip/hip_runtime.h>
#include <hip/hip_bf16.h>
#include <math.h>

// ---------------------------------------------------------------------------
// GNN NAS-cell forward for MI455X (gfx1250, wave32).
//  - 7x mixed(): edge gather + sigmoid gate + segment sum/max via fp32/u32
//    atomics (bandwidth-bound part, ~4 GB irregular traffic -> HBM-limited).
//  - Vc GEMM [N,256]x[256,64] and En GEMM [NE,192]x[192,64] use
//    V_WMMA_F32_16X16X4_F32 (exact fp32, matches reference precision; compute
//    is far below the memory roofline so fp32 WMMA is the right choice).
//  - BatchNorm (biased stats) + leaky_relu + residual epilogues.
// Workspace needed: ~295 MB.
// ---------------------------------------------------------------------------

typedef __attribute__((ext_vector_type(2))) float v2f;
typedef __attribute__((ext_vector_type(8))) float v8f;

#define DD 64
#define SLOPE 0.2f
#define BN_EPS 1e-5f

__device__ __forceinline__ float leaky(float x) { return x > 0.0f ? x : SLOPE * x; }
__device__ __forceinline__ float sigmoidf(float x) { return 1.0f / (1.0f + __expf(-x)); }

// Order-preserving float<->uint encoding so unsigned atomicMax == float max.
__device__ __forceinline__ unsigned fenc(float f) {
    unsigned u = __float_as_uint(f);
    return (u & 0x80000000u) ? ~u : (u | 0x80000000u);
}
__device__ __forceinline__ float fdec(unsigned u) {
    return __uint_as_float((u & 0x80000000u) ? (u & 0x7FFFFFFFu) : ~u);
}
#define FENC_NEG_INF 0x007FFFFFu  // fenc(-inf)

// ------------------------------- init ---------------------------------------
__global__ void k_init_deg_stats(float* deg, float* stats, int n) {
    int i = blockIdx.x * blockDim.x + threadIdx.x;
    if (i < n) deg[i] = 0.0f;
    if (i < 256) stats[i] = 0.0f;
}

__global__ void k_deg(const int* __restrict__ dst, float* __restrict__ deg, int ne) {
    int e = blockIdx.x * blockDim.x + threadIdx.x;
    if (e < ne) atomicAdd(&deg[dst[e]], 1.0f);
}

__global__ void k_init_sm(float* __restrict__ sum, unsigned* __restrict__ mxu, int n) {
    int i = blockIdx.x * blockDim.x + threadIdx.x;
    if (i < n) { sum[i] = 0.0f; mxu[i] = FENC_NEG_INF; }
}

// --------------------------- mixed(): scatter --------------------------------
// one thread per (edge, feature): m = x[src]*sigmoid(E); segsum + segmax
__global__ void k_edge_scatter(const float* __restrict__ x, const float* __restrict__ E,
                               const int* __restrict__ src, const int* __restrict__ dst,
                               float* __restrict__ sum, unsigned* __restrict__ mxu,
                               int total) {
    int i = blockIdx.x * blockDim.x + threadIdx.x;
    if (i >= total) return;
    int e = i >> 6, d = i & 63;
    float g = sigmoidf(E[i]);
    float m = x[src[e] * DD + d] * g;
    int o = dst[e] * DD + d;
    atomicAdd(&sum[o], m);
    atomicMax(&mxu[o], fenc(m));
}

// --------------------------- mixed(): per-node combine -----------------------
__global__ void k_accum_state(const float* __restrict__ x, const float* __restrict__ sum,
                              const unsigned* __restrict__ mxu, const float* __restrict__ deg,
                              const float* __restrict__ wrow, float* __restrict__ state,
                              int total, int add) {
    int i = blockIdx.x * blockDim.x + threadIdx.x;
    if (i >= total) return;
    int n = i >> 6;
    float w1 = wrow[1], w2 = wrow[2], w3 = wrow[3], w4 = wrow[4];
    float dg = deg[n];
    float s = sum[i];
    float mean = s / fmaxf(dg, 1.0f);
    float mx = (dg > 0.0f) ? fdec(mxu[i]) : 0.0f;
    float val = w1 * x[i] + w2 * mx + w3 * mean + w4 * s;
    state[i] = add ? (state[i] + val) : val;
}

// ------------------------ Vc GEMM: [N,256]x[256,64] --------------------------
// block = 128 threads (4 wave32s), one 16-row tile; wave w -> 16-col n-tile.
#define VC_K 256
#define VC_LDA 260  // pad: 260 % 64 banks = 4 -> 16 lanes hit 16 distinct banks
__global__ void k_vc_gemm(const float* __restrict__ s1, const float* __restrict__ s2,
                          const float* __restrict__ s3, const float* __restrict__ s4,
                          const float* __restrict__ Wc, const float* __restrict__ bc,
                          float* __restrict__ Vc) {
    __shared__ float lds[16 * VC_LDA];
    const int tid = threadIdx.x;
    const int rowbase = blockIdx.x * 16;

    // stage A-panel (16 x 256 concat of s1..s4) into LDS; uniform bounds
    for (int idx = tid; idx < 16 * VC_K; idx += 128) {
        int r = idx >> 8, c = idx & 255;
        int node = rowbase + r;
        const float* p = (c < 64) ? s1 : (c < 128) ? s2 : (c < 192) ? s3 : s4;
        lds[r * VC_LDA + c] = p[node * DD + (c & 63)];
    }
    __syncthreads();

    const int lane = tid & 31;
    const int wave = tid >> 5;
    const int n0 = wave * 16;
    const int half = lane & 15;
    const int lo = (lane >> 4) << 1;  // lanes 16-31 hold K+2, K+3 (ISA A/B layout)

    v8f acc = {};
    for (int k = 0; k < VC_K; k += 4) {
        v2f a, b;
        a.x = lds[half * VC_LDA + k + lo];
        a.y = lds[half * VC_LDA + k + lo + 1];
        b.x = Wc[(k + lo) * DD + n0 + half];
        b.y = Wc[(k + lo + 1) * DD + n0 + half];
        acc = __builtin_amdgcn_wmma_f32_16x16x4_f32(false, a, false, b, (short)0, acc,
                                                    false, false);
    }

    // C/D layout: vgpr j, lanes 0-15 -> M=j; lanes 16-31 -> M=j+8; N = lane&15
    const int mbase = (lane >> 4) * 8;
    const float bias = bc[n0 + half];
    for (int j = 0; j < 8; ++j) {
        int m = mbase + j;
        Vc[(rowbase + m) * DD + n0 + half] = acc[j] + bias;
    }
}

// ------------------------ En GEMM: [NE,192]x[192,64] -------------------------
// A row = [Vc[src[e]] | leaky(E[e]) | Vc[dst[e]]]; Vc (12.8MB) is L2-resident.
#define EN_K 192
#define EN_LDA 196
__global__ void k_en_gemm(const float* __restrict__ Vc, const float* __restrict__ E,
                          const int* __restrict__ src, const int* __restrict__ dst,
                          const float* __restrict__ Ws, const float* __restrict__ bS,
                          float* __restrict__ En) {
    __shared__ float lds[16 * EN_LDA];
    const int tid = threadIdx.x;
    const int rowbase = blockIdx.x * 16;

    for (int idx = tid; idx < 16 * EN_K; idx += 128) {
        int r = idx / EN_K, c = idx - r * EN_K;
        int e = rowbase + r;
        float v;
        if (c < 64)       v = Vc[src[e] * DD + c];
        else if (c < 128) v = leaky(E[e * DD + (c - 64)]);
        else              v = Vc[dst[e] * DD + (c - 128)];
        lds[r * EN_LDA + c] = v;
    }
    __syncthreads();

    const int lane = tid & 31;
    const int wave = tid >> 5;
    const int n0 = wave * 16;
    const int half = lane & 15;
    const int lo = (lane >> 4) << 1;

    v8f acc = {};
    for (int k = 0; k < EN_K; k += 4) {
        v2f a, b;
        a.x = lds[half * EN_LDA + k + lo];
        a.y = lds[half * EN_LDA + k + lo + 1];
        b.x = Ws[(k + lo) * DD + n0 + half];
        b.y = Ws[(k + lo + 1) * DD + n0 + half];
        acc = __builtin_amdgcn_wmma_f32_16x16x4_f32(false, a, false, b, (short)0, acc,
                                                    false, false);
    }

    const int mbase = (lane >> 4) * 8;
    const float bias = bS[n0 + half];
    for (int j = 0; j < 8; ++j) {
        int m = mbase + j;
        En[(size_t)(rowbase + m) * DD + n0 + half] = acc[j] + bias;
    }
}

// ------------------------- BN column statistics ------------------------------
__global__ void k_col_stats(const float* __restrict__ X, int rows,
                            float* __restrict__ colsum, float* __restrict__ colsq) {
    int t = blockIdx.x * blockDim.x + threadIdx.x;
    int d = t & 63;
    int rstart = t >> 6;
    int rstride = (gridDim.x * blockDim.x) >> 6;
    float s = 0.0f, q = 0.0f;
    for (int r = rstart; r < rows; r += rstride) {
        float v = X[(size_t)r * DD + d];
        s += v;
        q += v * v;
    }
    atomicAdd(&colsum[d], s);
    atomicAdd(&colsq[d], q);
}

// ----------------- BN + leaky_relu + residual epilogue -----------------------
__global__ void k_apply_bn(const float* __restrict__ X, const float* __restrict__ resid,
                           const float* __restrict__ colsum, const float* __restrict__ colsq,
                           const float* __restrict__ gamma, const float* __restrict__ beta,
                           float inv_rows, float* __restrict__ out, int total) {
    int i = blockIdx.x * blockDim.x + threadIdx.x;
    if (i >= total) return;
    int d = i & 63;
    float mu = colsum[d] * inv_rows;
    float var = colsq[d] * inv_rows - mu * mu;  // biased (jnp.var default)
    float y = (X[i] - mu) * rsqrtf(var + BN_EPS) * gamma[d] + beta[d];
    out[i] = leaky(y) + resid[i];
}

// ---------------------------------------------------------------------------
extern "C" void kernel_launch(void* const* d_in, const int* in_sizes, int n_in,
                              void* d_out, int out_size, void* d_ws, size_t ws_size,
                              hipStream_t stream) {
    const float* V      = (const float*)d_in[0];
    const float* E      = (const float*)d_in[1];
    const int*   src    = (const int*)d_in[2];
    const int*   dst    = (const int*)d_in[3];
    const float* weight = (const float*)d_in[4];  // [7,5]
    const float* W_cat  = (const float*)d_in[5];  // [256,64]
    const float* b_cat  = (const float*)d_in[6];
    const float* W_S    = (const float*)d_in[7];  // [192,64]
    const float* b_S    = (const float*)d_in[8];
    const float* gV     = (const float*)d_in[9];
    const float* bV     = (const float*)d_in[10];
    const float* gE     = (const float*)d_in[11];
    const float* bE     = (const float*)d_in[12];

    const int N  = in_sizes[0] / DD;  // 50000
    const int NE = in_sizes[1] / DD;  // 800000
    const int ND  = N * DD;           // 3.2M
    const int NED = NE * DD;          // 51.2M

    // --- workspace carve (256B aligned), ~295 MB total ---
    char* ws = (char*)d_ws;
    size_t off = 0;
    auto carve = [&](size_t bytes) {
        void* p = ws + off;
        off += (bytes + 255) & ~(size_t)255;
        return p;
    };
    float*    deg   = (float*)carve((size_t)N * 4);
    float*    s1    = (float*)carve((size_t)ND * 4);
    float*    s2    = (float*)carve((size_t)ND * 4);
    float*    s3    = (float*)carve((size_t)ND * 4);
    float*    s4    = (float*)carve((size_t)ND * 4);
    float*    sum   = (float*)carve((size_t)ND * 4);
    unsigned* mxu   = (unsigned*)carve((size_t)ND * 4);
    float*    Vc    = (float*)carve((size_t)ND * 4);
    float*    stats = (float*)carve(256 * 4);  // [sumV|sqV|sumE|sqE] x 64
    float*    En    = (float*)carve((size_t)NED * 4);
    (void)ws_size;

    float* sumV = stats + 0,  * sqV = stats + 64;
    float* sumE = stats + 128, * sqE = stats + 192;

    const int T = 256;
    dim3 blk(T);

    // degree + stats init
    k_init_deg_stats<<<dim3((N + T - 1) / T), blk, 0, stream>>>(deg, stats, N);
    k_deg<<<dim3((NE + T - 1) / T), blk, 0, stream>>>(dst, deg, NE);

    // one mixed() = init scratch, edge scatter, per-node combine
    auto mixed = [&](const float* x, int wr, float* state, int add) {
        k_init_sm<<<dim3((ND + T - 1) / T), blk, 0, stream>>>(sum, mxu, ND);
        k_edge_scatter<<<dim3((NED + T - 1) / T), blk, 0, stream>>>(x, E, src, dst, sum,
                                                                    mxu, NED);
        k_accum_state<<<dim3((ND + T - 1) / T), blk, 0, stream>>>(x, sum, mxu, deg,
                                                                  weight + wr * 5, state,
                                                                  ND, add);
    };

    // LINK: 1:[(0,0)] 2:[(0,1),(1,2)] 3:[(0,3),(2,4)] 4:[(0,5),(3,6)]
    mixed(V,  0, s1, 0);
    mixed(V,  1, s2, 0);  mixed(s1, 2, s2, 1);
    mixed(V,  3, s3, 0);  mixed(s2, 4, s3, 1);
    mixed(V,  5, s4, 0);  mixed(s3, 6, s4, 1);

    // Vc = concat(s1..s4) @ W_cat + b_cat   (fp32 WMMA 16x16x4)
    k_vc_gemm<<<dim3(N / 16), dim3(128), 0, stream>>>(s1, s2, s3, s4, W_cat, b_cat, Vc);

    // En = [Vc[src] | leaky(E) | Vc[dst]] @ W_S + b_S
    k_en_gemm<<<dim3(NE / 16), dim3(128), 0, stream>>>(Vc, E, src, dst, W_S, b_S, En);

    // BN stats (biased, over dim 0)
    k_col_stats<<<dim3(256), blk, 0, stream>>>(Vc, N, sumV, sqV);
    k_col_stats<<<dim3(256), blk, 0, stream>>>(En, NE, sumE, sqE);

    // outputs: Vo then Eo, concatenated flat
    float* Vo = (float*)d_out;
    float* Eo = (float*)d_out + (size_t)ND;
    k_apply_bn<<<dim3((ND + T - 1) / T), blk, 0, stream>>>(Vc, V, sumV, sqV, gV, bV,
                                                           1.0f / (float)N, Vo, ND);
    k_apply_bn<<<dim3((NED + T - 1) / T), blk, 0, stream>>>(En, E, sumE, sqE, gE, bE,
                                                            1.0f / (float)NE, Eo, NED);
    (void)out_size; (void)n_in;
}